// GATLayer_39616778338734
// MI455X (gfx1250) — compile-verified
//
#include <hip/hip_runtime.h>

typedef __attribute__((ext_vector_type(2))) float v2f;
typedef __attribute__((ext_vector_type(8))) float v8f;

#define IN_DIM_N 128
#define OUT_DIM  128
#define IN_DIM_E 32

// ---------------------------------------------------------------------------
// Ordered-int trick for float atomic max: works for all finite floats.
// Word always contains valid float bits. Init value must be -inf (0xFF800000).
// ---------------------------------------------------------------------------
__device__ __forceinline__ void atomicMaxF32(float* addr, float val) {
  if (val >= 0.0f) {
    atomicMax((int*)addr, __float_as_int(val));
  } else {
    atomicMin((unsigned int*)addr, __float_as_uint(val));
  }
}

// ---------------------------------------------------------------------------
// Kernel 1: zero output, init m = -inf, denom = 0
// ---------------------------------------------------------------------------
__global__ void gat_init_kernel(float* __restrict__ out,
                                unsigned int* __restrict__ m,
                                float* __restrict__ denom,
                                int nOut, int nNodes) {
  int i = blockIdx.x * blockDim.x + threadIdx.x;
  if (i < nOut) out[i] = 0.0f;
  if (i < nNodes) {
    m[i] = 0xFF800000u;  // -inf bit pattern
    denom[i] = 0.0f;
  }
}

// ---------------------------------------------------------------------------
// Kernel 2: z = nfeats @ W_fc^T via V_WMMA_F32_16X16X4_F32.
// One wave computes one 16x16 tile of z; 8 waves/block cover all 8 N-tiles,
// blockIdx.x selects the M-tile. K loop: 128 / 4 = 32 WMMAs per tile.
// ---------------------------------------------------------------------------
__global__ void gat_gemm_wmma_kernel(const float* __restrict__ A,   // nfeats [N,128]
                                     const float* __restrict__ W,   // W_fc   [128,128]
                                     float* __restrict__ z,         // [N,128]
                                     int nNodes) {
  const int lane  = threadIdx.x & 31;
  const int wave  = threadIdx.x >> 5;          // N-tile index: 0..7
  const int mtile = blockIdx.x;
  const int mlo   = lane & 15;
  const int khalf = (lane >> 4) << 1;          // 0 for lanes 0-15, 2 for 16-31

  int row = mtile * 16 + mlo;                  // A row (clamped to keep EXEC full)
  if (row >= nNodes) row = nNodes - 1;
  const int col = wave * 16 + mlo;             // W row == output dim

  const float* __restrict__ arow = A + (long)row * IN_DIM_N + khalf;
  const float* __restrict__ brow = W + (long)col * IN_DIM_N + khalf;

  v8f c = {};
#pragma unroll
  for (int k0 = 0; k0 < IN_DIM_N; k0 += 4) {
    v2f a = *(const v2f*)(arow + k0);          // A[M][k0+khalf .. +1]
    v2f b = *(const v2f*)(brow + k0);          // B[k0+khalf .. +1][N] = W[col][...]
    c = __builtin_amdgcn_wmma_f32_16x16x4_f32(
        /*neg_a=*/false, a, /*neg_b=*/false, b,
        /*c_mod=*/(short)0, c, /*reuse_a=*/false, /*reuse_b=*/false);
  }

  // C/D layout: VGPR r -> M = r + 8*(lane>=16), N = lane&15
  const int n0  = mtile * 16;
  const int mHi = (lane >> 4) << 3;
  const int o   = wave * 16 + mlo;
#pragma unroll
  for (int r = 0; r < 8; ++r) {
    int rr = n0 + mHi + r;
    if (rr < nNodes) z[(long)rr * OUT_DIM + o] = c[r];
  }
}

// ---------------------------------------------------------------------------
// Kernel 3: per-node attention scores s_src = z . a_src, s_dst = z . a_dst
// ---------------------------------------------------------------------------
__global__ void gat_score_kernel(const float* __restrict__ z,
                                 const float* __restrict__ Wattn,   // [288]
                                 float* __restrict__ s_src,
                                 float* __restrict__ s_dst,
                                 int nNodes) {
  int n = blockIdx.x * blockDim.x + threadIdx.x;
  if (n >= nNodes) return;
  const float* __restrict__ zr = z + (long)n * OUT_DIM;
  const float* __restrict__ a_src = Wattn;
  const float* __restrict__ a_dst = Wattn + OUT_DIM + IN_DIM_E;
  float ss = 0.0f, sd = 0.0f;
#pragma unroll 8
  for (int d = 0; d < OUT_DIM; ++d) {
    float zv = zr[d];
    ss += zv * a_src[d];
    sd += zv * a_dst[d];
  }
  s_src[n] = ss;
  s_dst[n] = sd;
}

// ---------------------------------------------------------------------------
// Kernel 4: per-edge score e = leaky_relu(s_src[src] + efeats.a_e + s_dst[dst])
//           + segment-max via float atomic max on m[dst]
// ---------------------------------------------------------------------------
__global__ void gat_edge_score_kernel(const float* __restrict__ efeats,  // [E,32]
                                      const float* __restrict__ Wattn,
                                      const int* __restrict__ src,
                                      const int* __restrict__ dst,
                                      const float* __restrict__ s_src,
                                      const float* __restrict__ s_dst,
                                      float* __restrict__ ebuf,
                                      float* __restrict__ m,
                                      int nEdges) {
  int i = blockIdx.x * blockDim.x + threadIdx.x;
  if (i >= nEdges) return;
  const float* __restrict__ er  = efeats + (long)i * IN_DIM_E;
  const float* __restrict__ a_e = Wattn + OUT_DIM;
  float se = 0.0f;
#pragma unroll
  for (int j = 0; j < IN_DIM_E; ++j) se += er[j] * a_e[j];
  float x  = s_src[src[i]] + se + s_dst[dst[i]];
  float ev = (x >= 0.0f) ? x : 0.01f * x;      // leaky_relu slope 0.01
  ebuf[i] = ev;
  atomicMaxF32(&m[dst[i]], ev);
}

// ---------------------------------------------------------------------------
// Kernel 5: w = exp(e - m[dst]); denom[dst] += w  (w overwrites e buffer)
// ---------------------------------------------------------------------------
__global__ void gat_edge_softmax_kernel(const int* __restrict__ dst,
                                        const float* __restrict__ m,
                                        float* __restrict__ ebuf,
                                        float* __restrict__ denom,
                                        int nEdges) {
  int i = blockIdx.x * blockDim.x + threadIdx.x;
  if (i >= nEdges) return;
  int d = dst[i];
  float w = expf(ebuf[i] - m[d]);
  ebuf[i] = w;
  atomicAdd(&denom[d], w);
}

// ---------------------------------------------------------------------------
// Kernel 6: h[dst] += (w/denom[dst]) * z[src].  One wave per edge; each lane
// owns 4 contiguous output dims (float4 gather from z, 4 atomic f32 adds).
// ---------------------------------------------------------------------------
__global__ void gat_aggregate_kernel(const float* __restrict__ z,
                                     const float* __restrict__ w,
                                     const float* __restrict__ denom,
                                     const int* __restrict__ src,
                                     const int* __restrict__ dst,
                                     float* __restrict__ out,
                                     int nEdges) {
  int edge = (int)((blockIdx.x * blockDim.x + threadIdx.x) >> 5);
  int lane = threadIdx.x & 31;
  if (edge >= nEdges) return;
  int s = src[edge];
  int d = dst[edge];
  float dn = denom[d];
  float alpha = w[edge] / (dn > 0.0f ? dn : 1.0f);
  const float4 zv = *(const float4*)(z + (long)s * OUT_DIM + lane * 4);
  float* __restrict__ o = out + (long)d * OUT_DIM + lane * 4;
  atomicAdd(o + 0, alpha * zv.x);
  atomicAdd(o + 1, alpha * zv.y);
  atomicAdd(o + 2, alpha * zv.z);
  atomicAdd(o + 3, alpha * zv.w);
}

// ---------------------------------------------------------------------------
extern "C" void kernel_launch(void* const* d_in, const int* in_sizes, int n_in,
                              void* d_out, int out_size, void* d_ws, size_t ws_size,
                              hipStream_t stream) {
  const float* nfeats = (const float*)d_in[0];
  const float* efeats = (const float*)d_in[1];
  const float* W_fc   = (const float*)d_in[2];
  const float* W_attn = (const float*)d_in[3];
  const int*   src    = (const int*)d_in[4];
  const int*   dst    = (const int*)d_in[5];
  float*       out    = (float*)d_out;

  const int nNodes = in_sizes[0] / IN_DIM_N;   // 50000
  const int nEdges = in_sizes[4];              // 800000
  const int nOut   = nNodes * OUT_DIM;

  // Workspace layout (floats)
  float* ws    = (float*)d_ws;
  float* z     = ws;                                  // nNodes*128
  float* s_src = z + (size_t)nNodes * OUT_DIM;        // nNodes
  float* s_dst = s_src + nNodes;                      // nNodes
  float* m     = s_dst + nNodes;                      // nNodes
  float* denom = m + nNodes;                          // nNodes
  float* ebuf  = denom + nNodes;                      // nEdges (e, then w)

  // 1) init out=0, m=-inf, denom=0
  gat_init_kernel<<<(nOut + 255) / 256, 256, 0, stream>>>(
      out, (unsigned int*)m, denom, nOut, nNodes);

  // 2) z = nfeats @ W_fc^T  (WMMA f32 16x16x4)
  const int mtiles = (nNodes + 15) / 16;
  gat_gemm_wmma_kernel<<<mtiles, 256, 0, stream>>>(nfeats, W_fc, z, nNodes);

  // 3) per-node scores
  gat_score_kernel<<<(nNodes + 255) / 256, 256, 0, stream>>>(
      z, W_attn, s_src, s_dst, nNodes);

  // 4) per-edge score + segment max
  gat_edge_score_kernel<<<(nEdges + 255) / 256, 256, 0, stream>>>(
      efeats, W_attn, src, dst, s_src, s_dst, ebuf, m, nEdges);

  // 5) per-edge exp + segment sum
  gat_edge_softmax_kernel<<<(nEdges + 255) / 256, 256, 0, stream>>>(
      dst, m, ebuf, denom, nEdges);

  // 6) weighted scatter-add (wave per edge)
  long aggThreads = (long)nEdges * 32;
  gat_aggregate_kernel<<<(int)((aggThreads + 255) / 256), 256, 0, stream>>>(
      z, ebuf, denom, src, dst, out, nEdges);
}